// AgreementRouting_11149735100930
// MI455X (gfx1250) — compile-verified
//
#include <hip/hip_runtime.h>

typedef __attribute__((ext_vector_type(2))) float v2f;
typedef __attribute__((ext_vector_type(8))) float v8f;

constexpr int kIC    = 1152;
constexpr int kOC    = 10;
constexpr int kD     = 16;
constexpr int kK     = kOC * kD;          // 160 flattened (j,d) columns
constexpr int kTiles = kIC / 16;          // 72 row tiles
constexpr int kChunk = kK / 4;            // 40 K-steps for wmma f32 16x16x4
constexpr int kThreads = 512;
constexpr int kWaves   = kThreads / 32;   // 16 wave32s
constexpr int kRowsPerWave = kIC / kWaves;// 72
constexpr int kUniformTiles = 4;          // 16 waves x 4 tiles = 64 tiles uniform
constexpr int kIter  = 3;

__global__ __launch_bounds__(kThreads)
void caps_routing_kernel(const float* __restrict__ u,
                         const float* __restrict__ bprior,
                         float* __restrict__ out)
{
    __shared__ float bb[kIC * kOC];      // routing logits, 46080 B
    __shared__ float s_lds[kOC * kD];    // weighted sum s, 640 B
    __shared__ float v_lds[kOC * kD];    // squashed v,     640 B

    const int tid   = threadIdx.x;
    const int wave  = tid >> 5;
    const int lane  = tid & 31;
    const int half  = lane >> 4;         // 0: K pair {0,1}, 1: K pair {2,3}
    const int m     = lane & 15;         // row (A) / col (B) index in tile
    const int batch = blockIdx.x;
    const float* uB = u + (size_t)batch * kIC * kK;

    for (int idx = tid; idx < kIC * kOC; idx += kThreads)
        bb[idx] = bprior[idx];
    __syncthreads();

    for (int iter = 0; iter <= kIter; ++iter) {
        // ---------- Phase 1 (iters 1..3): bb += U_flat @ blockdiag(v), WMMA f32 16x16x4
        if (iter > 0) {
            // Uniform part: wave handles tiles wave + 16*t, t = 0..3 (rows step 256).
            {
                const float* baseA = uB + (size_t)(wave * 16 + m) * kK;
                constexpr size_t tstep = (size_t)256 * kK;   // floats per tile step
                v8f acc[kUniformTiles] = {};
                for (int kc = 0; kc < kChunk; ++kc) {
                    const int c0   = kc * 4;
                    const int jp   = c0 >> 4;
                    const int dsel = (c0 & 15) + half * 2;
                    // broadcast LDS reads, branch-free select -> v_cndmask
                    const float bv0 = v_lds[jp * kD + dsel];
                    const float bv1 = v_lds[jp * kD + dsel + 1];
                    v2f Bm;
                    Bm.x = (m == jp) ? bv0 : 0.0f;
                    Bm.y = (m == jp) ? bv1 : 0.0f;
                    const int aoff = c0 + half * 2;
                    #pragma unroll
                    for (int t = 0; t < kUniformTiles; ++t) {
                        const v2f A = *(const v2f*)(baseA + (size_t)t * tstep + aoff);
                        acc[t] = __builtin_amdgcn_wmma_f32_16x16x4_f32(
                            false, A, false, Bm, (short)0, acc[t], false, false);
                    }
                }
                if (m < kOC) {
                    #pragma unroll
                    for (int t = 0; t < kUniformTiles; ++t) {
                        const int rowb = wave * 16 + t * 256 + half * 8;
                        #pragma unroll
                        for (int r = 0; r < 8; ++r)
                            bb[(rowb + r) * kOC + m] += acc[t][r];
                    }
                }
            }
            // Remainder: tiles 64..71, one per wave for waves 0..7.
            for (int tile = kUniformTiles * kWaves + wave; tile < kTiles; tile += kWaves) {
                const int i0 = tile * 16;
                const float* rowp = uB + (size_t)(i0 + m) * kK;
                v8f acc = {};
                for (int kc = 0; kc < kChunk; ++kc) {
                    const int c0   = kc * 4;
                    const int jp   = c0 >> 4;
                    const int dsel = (c0 & 15) + half * 2;
                    const float bv0 = v_lds[jp * kD + dsel];
                    const float bv1 = v_lds[jp * kD + dsel + 1];
                    v2f Bm;
                    Bm.x = (m == jp) ? bv0 : 0.0f;
                    Bm.y = (m == jp) ? bv1 : 0.0f;
                    const v2f A = *(const v2f*)(rowp + c0 + half * 2);
                    acc = __builtin_amdgcn_wmma_f32_16x16x4_f32(
                        false, A, false, Bm, (short)0, acc, false, false);
                }
                if (m < kOC) {
                    #pragma unroll
                    for (int r = 0; r < 8; ++r)
                        bb[(i0 + half * 8 + r) * kOC + m] += acc[r];
                }
            }
        }
        if (tid < kOC * kD) s_lds[tid] = 0.0f;
        __syncthreads();

        // ---------- Phase 2: per-row softmax over OC, accumulate s[j,d] = sum_i c*u
        {
            float part[5] = {0.f, 0.f, 0.f, 0.f, 0.f};
            const int c4 = lane * 4;          // aligned float4 block, single j-block
            const int cx = 128 + lane;        // tail column
            const int jA = lane >> 2;         // (c4 >> 4)
            const int jB = 8 + (lane >> 4);   // (cx >> 4)
            const int r0 = wave * kRowsPerWave;
            for (int i = r0; i < r0 + kRowsPerWave; ++i) {
                const float* brow = &bb[i * kOC];
                float mx = brow[0];
                #pragma unroll
                for (int j = 1; j < kOC; ++j) mx = fmaxf(mx, brow[j]);
                float sum = 0.0f;
                #pragma unroll
                for (int j = 0; j < kOC; ++j) sum += __expf(brow[j] - mx);
                const float inv = 1.0f / sum;
                const float cA = __expf(brow[jA] - mx) * inv;
                const float cB = __expf(brow[jB] - mx) * inv;
                const float* urow = uB + (size_t)i * kK;
                const float4 q = *(const float4*)(urow + c4);   // global_load_b128
                part[0] = fmaf(cA, q.x, part[0]);
                part[1] = fmaf(cA, q.y, part[1]);
                part[2] = fmaf(cA, q.z, part[2]);
                part[3] = fmaf(cA, q.w, part[3]);
                part[4] = fmaf(cB, urow[cx], part[4]);
            }
            #pragma unroll
            for (int e = 0; e < 4; ++e)
                atomicAdd(&s_lds[c4 + e], part[e]);   // ds_add_f32
            atomicAdd(&s_lds[cx], part[4]);
        }
        __syncthreads();

        // ---------- Squash: v = (|s|^2/(1+|s|^2)) * s / sqrt(|s|^2 + eps)
        if (tid < kOC * kD) {
            const int j = tid >> 4;
            float sq = 0.0f;
            #pragma unroll
            for (int dd = 0; dd < kD; ++dd) {
                const float sv = s_lds[j * kD + dd];
                sq = fmaf(sv, sv, sq);
            }
            const float scale = (sq / (1.0f + sq)) * rsqrtf(sq + 1e-8f);
            v_lds[tid] = scale * s_lds[tid];
        }
        __syncthreads();
    }

    if (tid < kOC * kD)
        out[(size_t)batch * kOC * kD + tid] = v_lds[tid];
}

extern "C" void kernel_launch(void* const* d_in, const int* in_sizes, int n_in,
                              void* d_out, int out_size, void* d_ws, size_t ws_size,
                              hipStream_t stream) {
    const float* u = (const float*)d_in[0];   // (B, IC, OC, D) fp32
    const float* b = (const float*)d_in[1];   // (IC, OC) fp32
    float* out = (float*)d_out;               // (B, OC, D) fp32
    const int B = in_sizes[0] / (kIC * kOC * kD);
    caps_routing_kernel<<<dim3(B), dim3(kThreads), 0, stream>>>(u, b, out);
}